// AggregationLayer_62319975465650
// MI455X (gfx1250) — compile-verified
//
#include <hip/hip_runtime.h>
#include <hip/hip_bf16.h>
#include <math.h>

// ---------------- problem constants ----------------
#define Bn      4
#define Hc      480
#define Wc      480
#define HW      (Hc * Wc)          // 230400
#define Kn      16
#define NINST   (Bn * Kn)          // 64

// ---------------- kernel-1 tiling ------------------
#define WGPI            32                         // workgroups per image
#define WAVES           8                          // waves per workgroup
#define THREADS         256
#define GROUPS_PER_WAVE (HW / 4 / (WGPI * WAVES))  // 225 groups of 4 pixels
#define PIX_PER_WG      (GROUPS_PER_WAVE * 4 * WAVES)  // 7200

// ---------------- d_out flat offsets (return order) ----------------
#define OFF_CLASS  ((size_t)0)
#define OFF_MASKS  ((size_t)64)
#define OFF_SAMPLE ((size_t)(64 + (size_t)NINST * HW))            // 14745664
#define OFF_QUAT   (OFF_SAMPLE + 64)                              // 14745728
#define OFF_SCALE  (OFF_QUAT + (size_t)NINST * 4)                 // 14745984
#define OFF_XY     (OFF_SCALE + (size_t)NINST * 3)                // 14746176
#define OFF_Z      (OFF_XY + (size_t)NINST * 2 * HW)              // 44237376

typedef float v2f __attribute__((ext_vector_type(2)));
typedef float v8f __attribute__((ext_vector_type(8)));

// ---------------------------------------------------------------------------
// Kernel 0: zero the 64 class-max slots in workspace (workspace is poisoned).
// ---------------------------------------------------------------------------
__global__ void agg_init_kernel(int* __restrict__ wsCls) {
    if (threadIdx.x < NINST) wsCls[threadIdx.x] = 0;
}

// ---------------------------------------------------------------------------
// Kernel 1: per-(instance, channel) sums via V_WMMA_F32_16X16X4_F32.
//   A (16x4, fp32)  : one-hot of instance id for 4 pixels; row M = instance k.
//   B (4x16, fp32)  : 4 pixels x 16 channels (quat0..3, scl0..2, z, count, 0..).
//   D (16x16, fp32) : accumulated sums, rows = instance, cols = channel.
// ISA VGPR layouts (CDNA5 ISA):
//   A: lanes 0-15 -> M=lane, {VGPR0,VGPR1}={K0,K1}; lanes 16-31 -> {K2,K3}.
//   B: lane carries N=lane%16;  lanes 0-15 -> {K0,K1}, lanes 16-31 -> {K2,K3}.
//   D: VGPR v holds M = v (lanes 0-15) / v+8 (lanes 16-31), N = lane%16.
//
// All lanes load unconditionally (cols >= 8 read the valid z plane) and the
// channel value is chosen with a v_cndmask afterwards -> no EXEC branches in
// the inner loop. The pixel pair {pA, pA+1} is fetched as one 8-byte load.
// ---------------------------------------------------------------------------
__global__ void __launch_bounds__(THREADS)
agg_reduce_kernel(const int*   __restrict__ cat,
                  const int*   __restrict__ lab,
                  const float* __restrict__ quat,
                  const float* __restrict__ scl,
                  const float* __restrict__ zin,
                  float* __restrict__ wsPartial,
                  int*   __restrict__ wsCls)
{
    __shared__ float red[WAVES][16][16];
    __shared__ int   cls[Kn];

    const int b    = blockIdx.x / WGPI;
    const int wg   = blockIdx.x % WGPI;
    const int tid  = threadIdx.x;
    const int wave = tid >> 5;
    const int lane = tid & 31;
    const int col  = lane & 15;      // A: row M; B: channel N
    const bool lo  = (lane < 16);    // low half-wave -> K pair {0,1}, else {2,3}

    // Channel source plane for the B matrix (per lane, loop-invariant).
    // Cols 8..15 alias the (valid) z plane; their loads are discarded below.
    const float* src;
    if (col < 4)      src = quat + (size_t)(b * 4 + col) * HW;
    else if (col < 7) src = scl  + (size_t)(b * 3 + (col - 4)) * HW;
    else              src = zin  + (size_t)b * HW;
    const bool  useMem = (col < 8);
    const float cConst = (col == 8) ? 1.0f : 0.0f;   // count channel / zeros

    const int* labp = lab + (size_t)b * HW;
    const int  kId  = col + 1;                       // instance id for row M

    const int gStart = (wg * WAVES + wave) * GROUPS_PER_WAVE;
    const int pStart = gStart * 4 + (lo ? 0 : 2);    // this lane's first pixel

    if (tid < Kn) cls[tid] = 0;

    v8f acc = {0.f, 0.f, 0.f, 0.f, 0.f, 0.f, 0.f, 0.f};

#pragma unroll 5
    for (int g = 0; g < GROUPS_PER_WAVE; ++g) {
        const int p = pStart + g * 4;                // even -> 8B aligned

        const int2   l2 = *(const int2*)  (labp + p);   // global_load_b64
        const float2 d2 = *(const float2*)(src  + p);   // global_load_b64

        v2f a, bm;
        a.x  = (l2.x == kId) ? 1.0f : 0.0f;
        a.y  = (l2.y == kId) ? 1.0f : 0.0f;
        bm.x = useMem ? d2.x : cConst;               // v_cndmask, no branch
        bm.y = useMem ? d2.y : cConst;

        // D = A(16x4) * B(4x16) + C   (exact fp32 accumulation)
        acc = __builtin_amdgcn_wmma_f32_16x16x4_f32(
                  /*neg_a=*/false, a, /*neg_b=*/false, bm,
                  /*c_mod=*/(short)0, acc,
                  /*reuse_a=*/false, /*reuse_b=*/false);
    }

    // Spill this wave's D tile to LDS.
#pragma unroll
    for (int v = 0; v < 8; ++v) {
        const int row = v + (lo ? 0 : 8);
        red[wave][row][col] = acc[v];
    }
    __syncthreads();

    // Deterministic cross-wave reduction (fixed order), one partial per WG.
    {
        const int row = tid >> 4;
        const int c   = tid & 15;
        float s = 0.0f;
#pragma unroll
        for (int w = 0; w < WAVES; ++w) s += red[w][row][c];
        wsPartial[((size_t)(b * WGPI + wg) * 256) + tid] = s;
    }

    // Integer class-id max over this WG's pixel slice (exact, order-free).
    {
        const int  pixBase = wg * PIX_PER_WG;
        const int* catp    = cat + (size_t)b * HW;
        for (int p = pixBase + tid; p < pixBase + PIX_PER_WG; p += THREADS) {
            const int l = labp[p];
            if (l > 0) atomicMax(&cls[l - 1], catp[p]);
        }
    }
    __syncthreads();
    if (tid < Kn) atomicMax(&wsCls[b * Kn + tid], cls[tid]);
}

// ---------------------------------------------------------------------------
// Kernel 2: fold the 32 per-image partials (fixed order) and emit the small
// outputs: class_ids, sample_ids, quat_agg (normalized), scales_agg, z_agg.
// ---------------------------------------------------------------------------
__global__ void agg_finalize_kernel(const float* __restrict__ wsPartial,
                                    const int*   __restrict__ wsCls,
                                    float* __restrict__ out)
{
    const int n = threadIdx.x;
    if (n >= NINST) return;
    const int b = n >> 4;
    const int k = n & 15;

    float s[9];
#pragma unroll
    for (int c = 0; c < 9; ++c) s[c] = 0.0f;
    for (int wg = 0; wg < WGPI; ++wg) {
        const float* p = wsPartial + ((size_t)(b * WGPI + wg) * 256) + k * 16;
#pragma unroll
        for (int c = 0; c < 9; ++c) s[c] += p[c];
    }

    const float cnt = s[8];
    const float inv = (cnt > 0.0f) ? (1.0f / cnt) : 0.0f;

    const float q0 = s[0] * inv, q1 = s[1] * inv, q2 = s[2] * inv, q3 = s[3] * inv;
    const float invn = 1.0f / sqrtf(q0 * q0 + q1 * q1 + q2 * q2 + q3 * q3);

    out[OFF_CLASS  + n] = (float)wsCls[n];
    out[OFF_SAMPLE + n] = (float)b;
    out[OFF_QUAT + (size_t)n * 4 + 0] = q0 * invn;
    out[OFF_QUAT + (size_t)n * 4 + 1] = q1 * invn;
    out[OFF_QUAT + (size_t)n * 4 + 2] = q2 * invn;
    out[OFF_QUAT + (size_t)n * 4 + 3] = q3 * invn;
    out[OFF_SCALE + (size_t)n * 3 + 0] = s[4] * inv;
    out[OFF_SCALE + (size_t)n * 3 + 1] = s[5] * inv;
    out[OFF_SCALE + (size_t)n * 3 + 2] = s[6] * inv;
    out[OFF_Z + n] = expf(s[7] * inv);
}

// ---------------------------------------------------------------------------
// Kernel 3: bandwidth kernel — write instance_masks [64,H,W] and
// xy_masked [64,2,H,W] with fully coalesced 128-bit loads/stores.
// Each thread owns 4 consecutive pixels of one image; label/xy read once,
// expanded to all 16 instances (48 float4 stores / thread, all streaming).
// This kernel carries the ~177 MB write traffic that sets the roofline.
// ---------------------------------------------------------------------------
__global__ void __launch_bounds__(256)
agg_expand_kernel(const int*   __restrict__ lab,
                  const float* __restrict__ xy,
                  float* __restrict__ out)
{
    const int gt = blockIdx.x * blockDim.x + threadIdx.x;   // 0 .. B*HW/4-1
    const int b  = gt / (HW / 4);
    const int q  = gt % (HW / 4);
    const size_t p = (size_t)q * 4;

    const int4   l4 = *(const int4*)  (lab + (size_t)b * HW + p);
    const float4 x4 = *(const float4*)(xy + ((size_t)b * 2 + 0) * HW + p);
    const float4 y4 = *(const float4*)(xy + ((size_t)b * 2 + 1) * HW + p);

    float* __restrict__ masks = out + OFF_MASKS;
    float* __restrict__ xym   = out + OFF_XY;

#pragma unroll
    for (int k = 0; k < Kn; ++k) {
        const int id = k + 1;
        float4 m;
        m.x = (l4.x == id) ? 1.0f : 0.0f;
        m.y = (l4.y == id) ? 1.0f : 0.0f;
        m.z = (l4.z == id) ? 1.0f : 0.0f;
        m.w = (l4.w == id) ? 1.0f : 0.0f;

        *(float4*)(masks + (size_t)(b * Kn + k) * HW + p) = m;

        float4 mx = {m.x * x4.x, m.y * x4.y, m.z * x4.z, m.w * x4.w};
        float4 my = {m.x * y4.x, m.y * y4.y, m.z * y4.z, m.w * y4.w};
        *(float4*)(xym + ((size_t)((b * Kn + k) * 2) + 0) * HW + p) = mx;
        *(float4*)(xym + ((size_t)((b * Kn + k) * 2) + 1) * HW + p) = my;
    }
}

// ---------------------------------------------------------------------------
extern "C" void kernel_launch(void* const* d_in, const int* in_sizes, int n_in,
                              void* d_out, int out_size, void* d_ws, size_t ws_size,
                              hipStream_t stream)
{
    const int*   cat  = (const int*)  d_in[0];   // cat_mask        [B,H,W]
    const int*   lab  = (const int*)  d_in[1];   // instance_labels [B,H,W]
    const float* quat = (const float*)d_in[2];   // [B,4,H,W]
    const float* scl  = (const float*)d_in[3];   // [B,3,H,W]
    const float* xy   = (const float*)d_in[4];   // [B,2,H,W]
    const float* zin  = (const float*)d_in[5];   // [B,H,W]
    float* out = (float*)d_out;

    float* wsPartial = (float*)d_ws;                                    // 128 KB
    int*   wsCls     = (int*)((char*)d_ws +
                              (size_t)Bn * WGPI * 256 * sizeof(float)); // 256 B

    hipLaunchKernelGGL(agg_init_kernel, dim3(1), dim3(64), 0, stream, wsCls);

    hipLaunchKernelGGL(agg_reduce_kernel, dim3(Bn * WGPI), dim3(THREADS), 0, stream,
                       cat, lab, quat, scl, zin, wsPartial, wsCls);

    hipLaunchKernelGGL(agg_finalize_kernel, dim3(1), dim3(NINST), 0, stream,
                       wsPartial, wsCls, out);

    hipLaunchKernelGGL(agg_expand_kernel, dim3((Bn * HW / 4) / 256), dim3(256),
                       0, stream, lab, xy, out);
}